// SparseMoE_22411139350728
// MI455X (gfx1250) — compile-verified
//
#include <hip/hip_runtime.h>
#include <cstdint>

#define HIDDEN 2048
#define EXPERTS 8
#define NTOK    8192   // BATCH * SEQ

typedef __attribute__((ext_vector_type(2))) float v2f;
typedef __attribute__((ext_vector_type(8))) float v8f;
typedef int v4i __attribute__((vector_size(16)));   // 16B chunk for async LDS DMA

#if defined(__has_builtin)
#if __has_builtin(__builtin_amdgcn_global_load_async_to_lds_b128)
#define ASYNC_LDS 1
#endif
#endif

#if defined(ASYNC_LDS)
#if __has_builtin(__builtin_amdgcn_s_wait_asynccnt)
#define WAIT_ASYNC(n) __builtin_amdgcn_s_wait_asynccnt(n)
#else
#define WAIT_ASYNC(n) asm volatile("s_wait_asynccnt %0" ::"i"(n) : "memory")
#endif
#endif

// ---------------------------------------------------------------------------
// Router: one wave (32 lanes) per token. logits = x @ gate_w^T + b, softmax,
// top-2, normalized weights; emits logits/weights/sel/expert_mask and a dense
// combine[N][E] table for the expert GEMM.
// ---------------------------------------------------------------------------
__global__ __launch_bounds__(256) void moe_router_gfx1250(
    const float* __restrict__ x, const float* __restrict__ gw,
    const float* __restrict__ gb, float* __restrict__ logits,
    float* __restrict__ rweights, int* __restrict__ sel,
    int* __restrict__ mask, float* __restrict__ combine)
{
    const int lane = threadIdx.x & 31;
    const int wave = threadIdx.x >> 5;
    const int n    = blockIdx.x * 8 + wave;

    float acc[EXPERTS];
#pragma unroll
    for (int e = 0; e < EXPERTS; ++e) acc[e] = 0.f;

    const float* xp = x + (size_t)n * HIDDEN;
    for (int d = lane; d < HIDDEN; d += 32) {
        const float xv = xp[d];
#pragma unroll
        for (int e = 0; e < EXPERTS; ++e)
            acc[e] = fmaf(xv, gw[e * HIDDEN + d], acc[e]);
    }
#pragma unroll
    for (int e = 0; e < EXPERTS; ++e) {
#pragma unroll
        for (int off = 16; off >= 1; off >>= 1)
            acc[e] += __shfl_xor(acc[e], off, 32);
    }

    if (lane == 0) {
        float lg[EXPERTS], p[EXPERTS];
        float mx = -3.402823e38f;
#pragma unroll
        for (int e = 0; e < EXPERTS; ++e) {
            lg[e] = acc[e] + gb[e];
            logits[n * EXPERTS + e] = lg[e];
            mx = fmaxf(mx, lg[e]);
        }
        float s = 0.f;
#pragma unroll
        for (int e = 0; e < EXPERTS; ++e) { p[e] = __expf(lg[e] - mx); s += p[e]; }
        const float invs = 1.f / s;
#pragma unroll
        for (int e = 0; e < EXPERTS; ++e) p[e] *= invs;

        int i0 = 0;
#pragma unroll
        for (int e = 1; e < EXPERTS; ++e) if (p[e] > p[i0]) i0 = e;
        int i1 = (i0 == 0) ? 1 : 0;
#pragma unroll
        for (int e = 0; e < EXPERTS; ++e)
            if (e != i0 && p[e] > p[i1]) i1 = e;

        float w0 = p[i0], w1 = p[i1];
        const float invw = 1.f / (w0 + w1);
        w0 *= invw; w1 *= invw;

        rweights[n * 2 + 0] = w0;
        rweights[n * 2 + 1] = w1;
        sel[n * 2 + 0] = i0;
        sel[n * 2 + 1] = i1;
#pragma unroll
        for (int e = 0; e < EXPERTS; ++e) {
            combine[(size_t)n * EXPERTS + e] = (e == i0) ? w0 : ((e == i1) ? w1 : 0.f);
            mask[(e * 2 + 0) * NTOK + n] = (i0 == e) ? 1 : 0;
            mask[(e * 2 + 1) * NTOK + n] = (i1 == e) ? 1 : 0;
        }
    }
}

// ---------------------------------------------------------------------------
// Expert GEMM: out[n,f] = sum_e (c[n,e]*x[n,:]) . W_e[f,:] + (combine @ b).
// Block: 512 threads (16 waves), tile 128 tokens x 32 features; one 16x16
// WMMA accumulator per wave summed over all experts (A fragment scaled by
// combine weight in registers). K chunked by 16 through double-buffered LDS
// staged with GLOBAL_LOAD_ASYNC_TO_LDS_B128 (ASYNCcnt) when available.
// ---------------------------------------------------------------------------
#define MT  128
#define NT  32
#define KC  16
#define ALD 20   // padded row stride: 16B aligned rows, conflict-free mod 64
#define BLD 20
#define NCH (HIDDEN / KC)   // 128 chunks

__global__ __launch_bounds__(512) void moe_expert_wmma_gfx1250(
    const float* __restrict__ x, const float* __restrict__ ew,
    const float* __restrict__ eb, const float* __restrict__ combine,
    float* __restrict__ out)
{
    __shared__ __align__(16) float As[2][MT * ALD];          // 2 x 10 KB
    __shared__ __align__(16) float Bs[2][EXPERTS][32 * BLD]; // 2 x 20 KB

    const int tid  = threadIdx.x;
    const int lane = tid & 31;
    const int wave = tid >> 5;          // 0..15
    const int mT   = wave >> 1;         // 0..7  token sub-tile
    const int nT   = wave & 1;          // 0..1  feature sub-tile
    const int f0   = blockIdx.x * NT;
    const int m0   = blockIdx.y * MT;

    v8f acc = {0.f, 0.f, 0.f, 0.f, 0.f, 0.f, 0.f, 0.f};

    // per-lane combine weights for this wave's 16 token rows (A-frag rows)
    const float* crow = combine + (size_t)(m0 + mT * 16 + (lane & 15)) * EXPERTS;
    const float4 c0 = *(const float4*)crow;
    const float4 c1 = *(const float4*)(crow + 4);
    const float ce[EXPERTS] = {c0.x, c0.y, c0.z, c0.w, c1.x, c1.y, c1.z, c1.w};

    // staging geometry (one 16B chunk per lane-op)
    const int arow = tid >> 2;            // 0..127
    const int acol = (tid & 3) << 2;      // 0,4,8,12

    auto stage = [&](int buf, int kc) {
        {   // A tile: x[m0..m0+127][kc..kc+15]  (unscaled)
            const float* g = x + (size_t)(m0 + arow) * HIDDEN + kc + acol;
            float* l = &As[buf][arow * ALD + acol];
#if defined(ASYNC_LDS)
            __builtin_amdgcn_global_load_async_to_lds_b128(
                (__attribute__((address_space(1))) v4i*)g,
                (__attribute__((address_space(3))) v4i*)l, 0, 0);
#else
            const float4 v = *(const float4*)g;
            v2f lo; lo.x = v.x; lo.y = v.y;
            v2f hi; hi.x = v.z; hi.y = v.w;
            *(v2f*)l = lo; *(v2f*)(l + 2) = hi;
#endif
        }
#pragma unroll
        for (int r = 0; r < 2; ++r) {   // B tiles: W_e[f0..f0+31][kc..kc+15], all 8 experts
            const int id   = tid + (r << 9);
            const int row  = id >> 2;          // 0..255
            const int e    = row >> 5;
            const int frow = row & 31;
            const int col  = (id & 3) << 2;
            const float* g = ew + ((size_t)e * HIDDEN + f0 + frow) * HIDDEN + kc + col;
            float* l = &Bs[buf][e][frow * BLD + col];
            if (r == 0 && kc + 4 * KC < HIDDEN)
                __builtin_prefetch(g + 4 * KC, 0, 1);   // global_prefetch_b8
#if defined(ASYNC_LDS)
            __builtin_amdgcn_global_load_async_to_lds_b128(
                (__attribute__((address_space(1))) v4i*)g,
                (__attribute__((address_space(3))) v4i*)l, 0, 0);
#else
            const float4 v = *(const float4*)g;
            v2f lo; lo.x = v.x; lo.y = v.y;
            v2f hi; hi.x = v.z; hi.y = v.w;
            *(v2f*)l = lo; *(v2f*)(l + 2) = hi;
#endif
        }
    };

    const int koff = (lane >> 4) << 1;   // lanes 16..31 hold K+2,K+3 (ISA 7.12.2)
    const int afr  = (mT * 16 + (lane & 15)) * ALD + koff;
    const int bfr  = (nT * 16 + (lane & 15)) * BLD + koff;

    auto compute = [&](int buf) {
        const float* ap = &As[buf][afr];
#pragma unroll
        for (int k = 0; k < KC; k += 4) {
            const v2f a = *(const v2f*)(ap + k);     // raw A frag, shared across experts
#pragma unroll
            for (int e = 0; e < EXPERTS; ++e) {
                const v2f ae = a * ce[e];            // combine-weight scaling (VALU, co-exec)
                const v2f b  = *(const v2f*)(&Bs[buf][e][bfr + k]);
                acc = __builtin_amdgcn_wmma_f32_16x16x4_f32(
                    false, ae, false, b, (short)0, acc, false, false);
            }
        }
    };

    // double-buffered pipeline: stage chunk c+1 while computing chunk c
    stage(0, 0);
    for (int c = 0; c < NCH; ++c) {
        const int  buf     = c & 1;
        const bool hasNext = (c + 1) < NCH;
        if (hasNext) stage(buf ^ 1, (c + 1) * KC);
#if defined(ASYNC_LDS)
        if (hasNext) WAIT_ASYNC(3);   // 3 async ops in flight for next chunk
        else         WAIT_ASYNC(0);
#endif
        __syncthreads();
        compute(buf);
        __syncthreads();
    }

    // epilogue: combine-weighted bias + store (C layout: VGPR v -> M=v / v+8)
    const int fcol  = f0 + nT * 16 + (lane & 15);
    const int mbase = m0 + mT * 16 + ((lane >> 4) << 3);
#pragma unroll
    for (int v = 0; v < 8; ++v) {
        const int m = mbase + v;
        float bias = 0.f;
#pragma unroll
        for (int e = 0; e < EXPERTS; ++e)
            bias = fmaf(combine[(size_t)m * EXPERTS + e], eb[e * HIDDEN + fcol], bias);
        out[(size_t)m * HIDDEN + fcol] = acc[v] + bias;
    }
}

// ---------------------------------------------------------------------------
extern "C" void kernel_launch(void* const* d_in, const int* in_sizes, int n_in,
                              void* d_out, int out_size, void* d_ws, size_t ws_size,
                              hipStream_t stream)
{
    (void)in_sizes; (void)n_in; (void)out_size; (void)ws_size;
    const float* x  = (const float*)d_in[0];
    const float* gw = (const float*)d_in[1];
    const float* gb = (const float*)d_in[2];
    const float* ew = (const float*)d_in[3];
    const float* eb = (const float*)d_in[4];

    float* out    = (float*)d_out;
    float* final_ = out;                                  // (B,S,H)
    float* logits = final_ + (size_t)NTOK * HIDDEN;       // (N,E)
    float* rw     = logits + (size_t)NTOK * EXPERTS;      // (N,K)
    int*   sel    = (int*)(rw + (size_t)NTOK * 2);        // (N,K) int32
    int*   mask   = sel + (size_t)NTOK * 2;               // (E,K,N) int32

    float* combine = (float*)d_ws;                        // N*E floats

    moe_router_gfx1250<<<NTOK / 8, 256, 0, stream>>>(
        x, gw, gb, logits, rw, sel, mask, combine);

    dim3 grid(HIDDEN / NT, NTOK / MT);   // (64, 64)
    moe_expert_wmma_gfx1250<<<grid, 512, 0, stream>>>(
        x, ew, eb, combine, final_);
}